// Spatialize_78305843740704
// MI455X (gfx1250) — compile-verified
//
#include <hip/hip_runtime.h>

typedef float v2f __attribute__((ext_vector_type(2)));
typedef float v8f __attribute__((ext_vector_type(8)));

constexpr int T_LEN   = 150000;
constexpr int N_TAPS  = 10000;
constexpr int OUT_LEN = 125000;
constexpr int START   = 12500;
constexpr int NCH     = 2;

constexpr int KB      = 1024;              // taps per K-block staged in LDS
constexpr int K_TOT   = 10240;             // padded K sweep: k in [-16, 10224)
constexpr int NKB     = K_TOT / KB;        // 10
constexpr int WAVES   = 8;                 // waves per workgroup
constexpr int TPW     = 4;                 // 16x16 output tiles per wave (share B)
constexpr int TILE_W  = 256;               // outputs per tile
constexpr int WAVE_OUT= TPW * TILE_W;      // 1024 outputs per wave
constexpr int WG_OUT  = WAVES * WAVE_OUT;  // 8192 outputs per workgroup
constexpr int ASPAN   = WG_OUT - TILE_W + 240;      // 8176: max (tb_off + 16r)
constexpr int XS_LEN  = KB + ASPAN + 24;   // 9224 logical x floats (needs KB+8175+1)
constexpr int XS_PHYS = XS_LEN + XS_LEN / 64 + 2;   // skew: phys(j) = j + (j>>6)

#define WMMA_F32(A, B, C) \
  __builtin_amdgcn_wmma_f32_16x16x4_f32(false, (A), false, (B), (short)0, (C), false, false)

__global__ __launch_bounds__(WAVES * 32)
void spatialize_fir_wmma(const float* __restrict__ x,
                         const float* __restrict__ ir,
                         float* __restrict__ out) {
  __shared__ float x_s[XS_PHYS];     // holds x REVERSED: x_s[j] = x[xhi - j]
  __shared__ float irL_s[KB + 16];
  __shared__ float irR_s[KB + 16];

  const int tid  = threadIdx.x;
  const int lane = tid & 31;
  const int wave = tid >> 5;
  const int half = lane >> 4;   // 0: lanes 0-15, 1: lanes 16-31
  const int n16  = lane & 15;

  const int b       = blockIdx.y;
  const int wg_base = blockIdx.x * WG_OUT;

  const float* xb = x + (long)b * T_LEN;

  v8f accL[TPW] = {};
  v8f accR[TPW] = {};

  // Reversed-x A indexing: logical j of tile t's A VGPR0 element is
  //   j0 = ASPAN - wave*WAVE_OUT - 256*t - 16*r + 4*ks + 2*half   (even)
  // and VGPR1 sits at j0+1 (ascending!), so the ds pair loads in register order.
  // Anchor at tile 3 (smallest j); tiles then sit at +260*(3-t) after skew.
  const int aBaseS = ASPAN - wave * WAVE_OUT - 16 * n16 + 2 * half - 256 * (TPW - 1);
  // B (4x16): lane col n=n16; VGPR0 = ir_pad[k0 + 2*half + n], VGPR1 = +1.
  const int bBase  = 2 * half + n16;

  for (int kb = 0; kb < NKB; ++kb) {
    const int K0g = kb * KB - 16;                    // global k at block start
    const int xhi = wg_base + START + ASPAN - K0g;   // x index of x_s[0] (reversed)

    __syncthreads();  // previous block's compute done before LDS overwrite

    for (int j = tid; j < XS_LEN; j += WAVES * 32) {
      int g = xhi - j;
      float v = (g >= 0 && g < T_LEN) ? xb[g] : 0.0f;  // provably in-range; cheap guard
      x_s[j + (j >> 6)] = v;
    }
    for (int j = tid; j < KB + 16; j += WAVES * 32) {
      int m = K0g + j;
      bool ok = (m >= 0) && (m < N_TAPS);
      irL_s[j] = ok ? ir[2 * m + 0] : 0.0f;
      irR_s[j] = ok ? ir[2 * m + 1] : 0.0f;
    }
    __syncthreads();

#pragma unroll 4
    for (int ks = 0; ks < KB / 4; ++ks) {
      // One B pair per channel feeds all 4 M-tiles.
      const int q = bBase + 4 * ks;
      v2f bl = { irL_s[q], irL_s[q + 1] };
      v2f br = { irR_s[q], irR_s[q + 1] };

      // p even -> (p, p+1) share a skew block; 256-float tile step -> +260 skewed.
      const int p = aBaseS + 4 * ks;
      const int s = p + (p >> 6);
      v2f a3 = { x_s[s],       x_s[s + 1]   };   // output tile 3
      v2f a2 = { x_s[s + 260], x_s[s + 261] };   // output tile 2
      v2f a1 = { x_s[s + 520], x_s[s + 521] };   // output tile 1
      v2f a0 = { x_s[s + 780], x_s[s + 781] };   // output tile 0

      accL[0] = WMMA_F32(a0, bl, accL[0]);
      accR[0] = WMMA_F32(a0, br, accR[0]);
      accL[1] = WMMA_F32(a1, bl, accL[1]);
      accR[1] = WMMA_F32(a1, br, accR[1]);
      accL[2] = WMMA_F32(a2, bl, accL[2]);
      accR[2] = WMMA_F32(a2, br, accR[2]);
      accL[3] = WMMA_F32(a3, bl, accL[3]);
      accR[3] = WMMA_F32(a3, br, accR[3]);
    }
  }

  // D layout: lane col n=n16; VGPR v holds rows M = v + 8*half. sample = tb + 16*M + n.
  float* oL = out + ((long)b * NCH + 0) * OUT_LEN;
  float* oR = oL + OUT_LEN;
#pragma unroll
  for (int t = 0; t < TPW; ++t) {
    const int tb = wg_base + wave * WAVE_OUT + t * TILE_W;
#pragma unroll
    for (int v = 0; v < 8; ++v) {
      int i = tb + 16 * (v + 8 * half) + n16;
      if (i < OUT_LEN) {
        oL[i] = accL[t][v];
        oR[i] = accR[t][v];
      }
    }
  }
}

extern "C" void kernel_launch(void* const* d_in, const int* in_sizes, int n_in,
                              void* d_out, int out_size, void* d_ws, size_t ws_size,
                              hipStream_t stream) {
  (void)in_sizes; (void)n_in; (void)d_ws; (void)ws_size; (void)out_size;
  const float* x  = (const float*)d_in[0];   // (64, 150000) f32
  const float* ir = (const float*)d_in[1];   // (10000, 2)   f32
  float* out      = (float*)d_out;           // (64, 2, 125000) f32

  dim3 grid((OUT_LEN + WG_OUT - 1) / WG_OUT, 64);  // (16, 64)
  dim3 block(WAVES * 32);                          // 256 threads = 8 wave32
  spatialize_fir_wmma<<<grid, block, 0, stream>>>(x, ir, out);
}